// LogEig_11605001634479
// MI455X (gfx1250) — compile-verified
//
#include <hip/hip_runtime.h>
#include <math.h>

typedef float v2f __attribute__((ext_vector_type(2)));
typedef float v8f __attribute__((ext_vector_type(8)));

// Load full 32x32 operand fragments for V_WMMA_F32_16X16X4_F32.
// A-layout: element [M = lane&15 (+16*I)][K = 4c + 2*(lane>>4) + j], j = vector slot.
// Matrices here are always symmetric, so the same fragments serve as B operands.
__device__ __forceinline__ void load_frags(const float* M, int n, int hi, v2f F[2][8]) {
#pragma unroll
  for (int I = 0; I < 2; ++I)
#pragma unroll
    for (int c = 0; c < 8; ++c) {
      const float* p = M + (16 * I + n) * 32 + 4 * c + 2 * hi;
      F[I][c] = *(const v2f*)p;  // ds_load_b64
    }
}

// 32x32 f32 matmul on one wave via 4 tiles x 8 K-chunks of v_wmma_f32_16x16x4_f32.
// MODE 0: dst = A*B      MODE 1: dst = (3I - A*B)/2      MODE 2: dst = alpha*I + A*B
// In-place (dst==A or dst==B) is safe: all operand loads precede all stores,
// and LDS ops from one wave complete in order.
template <int MODE>
__device__ __forceinline__ void mm32(float* dst, const float* A, const float* B,
                                     int lane, float alpha) {
  const int n = lane & 15, hi = lane >> 4;
  v2f FA[2][8], FB[2][8];
  load_frags(A, n, hi, FA);
  load_frags(B, n, hi, FB);
#pragma unroll
  for (int I = 0; I < 2; ++I)
#pragma unroll
    for (int J = 0; J < 2; ++J) {
      v8f acc = {};
#pragma unroll
      for (int c = 0; c < 8; ++c)
        acc = __builtin_amdgcn_wmma_f32_16x16x4_f32(
            false, FA[I][c], false, FB[J][c], (short)0, acc, false, false);
#pragma unroll
      for (int v = 0; v < 8; ++v) {
        const int row = 16 * I + v + 8 * hi;  // C/D layout: VGPR v -> M = v + 8*hi
        const int col = 16 * J + n;           //             N = lane & 15
        float val = acc[v];
        if (MODE == 1) val = ((row == col) ? 3.0f - val : -val) * 0.5f;
        if (MODE == 2) val = val + ((row == col) ? alpha : 0.0f);
        dst[row * 32 + col] = val;
      }
    }
}

__global__ __launch_bounds__(128) void logm32_kernel(const float* __restrict__ P,
                                                     float* __restrict__ out, int nmat) {
  __shared__ float sY[4][1024];
  __shared__ float sZ[4][1024];
  __shared__ float sT[4][1024];
  const int lane = threadIdx.x & 31;
  const int wid = threadIdx.x >> 5;
  const int mat = blockIdx.x * 4 + wid;
  if (mat >= nmat) return;  // wave-uniform: EXEC stays all-ones for WMMA
  float* Y = sY[wid];
  float* Z = sZ[wid];
  float* T = sT[wid];

  const float4* Pg = (const float4*)(P + (size_t)mat * 1024);
  float4* Og = (float4*)(out + (size_t)mat * 1024);

  // Load P into LDS (b128 global loads), accumulate Frobenius norm^2.
  float ssq = 0.f;
#pragma unroll
  for (int i = 0; i < 8; ++i) {
    float4 v = Pg[lane + 32 * i];
    ssq += v.x * v.x + v.y * v.y + v.z * v.z + v.w * v.w;
    ((float4*)Y)[lane + 32 * i] = v;
  }
#pragma unroll
  for (int m = 16; m >= 1; m >>= 1) ssq += __shfl_xor(ssq, m, 32);
  const float c = sqrtf(ssq);   // >= lambda_max(P); spec(P/c) in (0,1]
  const float rc = 1.0f / c;
#pragma unroll
  for (int i = 0; i < 8; ++i) {
    float4 v = ((float4*)Y)[lane + 32 * i];
    v.x *= rc; v.y *= rc; v.z *= rc; v.w *= rc;
    ((float4*)Y)[lane + 32 * i] = v;
  }

  // 5 stages of coupled Newton-Schulz sqrt: Y -> Y^(1/2) each stage.
  const int iters[5] = {10, 8, 6, 5, 4};
#pragma unroll
  for (int s = 0; s < 5; ++s) {
#pragma unroll 1
    for (int i = 0; i < 32; ++i) Z[lane * 32 + i] = (lane == i) ? 1.0f : 0.0f;
#pragma unroll 1
    for (int it = 0; it < iters[s]; ++it) {
      mm32<1>(T, Z, Y, lane, 0.f);  // T = (3I - Z*Y)/2
      mm32<0>(Y, Y, T, lane, 0.f);  // Y = Y*T
      mm32<0>(Z, T, Z, lane, 0.f);  // Z = T*Z
    }
  }

  // E = X - I  (each lane owns diagonal element of its row)
  Y[lane * 33] -= 1.0f;

  // 8-term Taylor log(I+E) via Horner: S = c7*I + c8*E; S = c_n*I + E*S; L = E*S
#pragma unroll 1
  for (int i = 0; i < 32; ++i)
    Z[lane * 32 + i] = (-1.0f / 8.0f) * Y[lane * 32 + i] + ((lane == i) ? (1.0f / 7.0f) : 0.0f);
#pragma unroll
  for (int n2 = 0; n2 < 6; ++n2) {
    const int nn = 6 - n2;                              // n = 6..1
    const float alpha = ((nn & 1) ? 1.0f : -1.0f) / (float)nn;
    mm32<2>(Z, Y, Z, lane, alpha);
  }
  mm32<0>(T, Y, Z, lane, 0.f);  // L = log(P/c) / 32

  // out = 32*L + log(c)*I
  const float lc = logf(c);
#pragma unroll
  for (int i = 0; i < 8; ++i) {
    float4 v = ((float4*)T)[lane + 32 * i];
    const int f = (lane + 32 * i) * 4;
    float o[4] = {v.x, v.y, v.z, v.w};
#pragma unroll
    for (int k = 0; k < 4; ++k) {
      const int row = (f + k) >> 5, col = (f + k) & 31;
      o[k] = 32.0f * o[k] + ((row == col) ? lc : 0.0f);
    }
    Og[lane + 32 * i] = make_float4(o[0], o[1], o[2], o[3]);
  }
}

extern "C" void kernel_launch(void* const* d_in, const int* in_sizes, int n_in,
                              void* d_out, int out_size, void* d_ws, size_t ws_size,
                              hipStream_t stream) {
  const float* P = (const float*)d_in[0];
  float* out = (float*)d_out;
  const int nmat = in_sizes[0] / 1024;        // 32*32 floats per matrix
  const int nblocks = (nmat + 3) / 4;         // 4 waves (matrices) per 128-thread block
  hipLaunchKernelGGL(logm32_kernel, dim3(nblocks), dim3(128), 0, stream, P, out, nmat);
}